// Transolver_15599321219569
// MI455X (gfx1250) — compile-verified
//
#include <hip/hip_runtime.h>
#include <math.h>

#define HIDDIM 256
#define HEADS  8
#define DHEAD  32
#define SLICES 32
#define NLAYERS 5
#define BGRAPH 16
#define PREHID 512
#define OUTD   4

typedef __attribute__((ext_vector_type(2))) float v2f;
typedef __attribute__((ext_vector_type(8))) float v8f;

__device__ __forceinline__ float gelu_tanh(float x) {
  const float c0 = 0.7978845608028654f;  // sqrt(2/pi)
  return 0.5f * x * (1.0f + tanhf(c0 * (x + 0.044715f * x * x * x)));
}

__device__ __forceinline__ float wave_sum(float v) {
#pragma unroll
  for (int m = 16; m >= 1; m >>= 1) v += __shfl_xor(v, m, 32);
  return v;
}
__device__ __forceinline__ float wave_max(float v) {
#pragma unroll
  for (int m = 16; m >= 1; m >>= 1) v = fmaxf(v, __shfl_xor(v, m, 32));
  return v;
}

// -------------------------------------------------------------------------
// fp32 WMMA GEMM:  Out[nrows,256] = act(A[nrows,K] @ Bw[K,256] + bias (+bias2))
//                  (+ Res)   ; act: 0=none, 1=gelu
// Block = 256 threads (8 waves); each block does 128 rows; each wave 16 rows.
// B is staged per 32-wide K-chunk into LDS in WMMA B-fragment layout, with a
// register-level software pipeline: while chunk k is computed from LDS, chunk
// k+1 is prefetched from global into registers. Within a k-step, all 16 B
// fragments are register-staged before the 16-WMMA burst so DS latency is
// paid once per k-step instead of per WMMA pair.
// -------------------------------------------------------------------------
__global__ __launch_bounds__(256) void wmma_gemm_n256(
    const float* __restrict__ A, const float* __restrict__ Bw,
    const float* __restrict__ bias, const float* __restrict__ bias2,
    const float* Res, float* Out, int nrows, int K, int act) {
  __shared__ float2 lB[8][16][32];  // [kstep][ntile][lane] = 32 KB

  const int tid = threadIdx.x;
  const int lane = tid & 31;
  const int wv = tid >> 5;
  const int r0 = blockIdx.x * 128 + wv * 16;

  v8f acc[16] = {};

  // A fragment source: lane l holds row (l&15), k-pair offset 0 or 2
  int arow = r0 + (lane & 15);
  if (arow > nrows - 1) arow = nrows - 1;
  const float* Arow = A + (size_t)arow * K + ((lane >> 4) << 1);

  // Each thread stages 16 fragment elements per chunk. For element i:
  //   idx8 = i*8 + wv;  nt = idx8 & 15;  ks = idx8 >> 4
  //   k = kc + ks*4 + (lane>=16 ? 2 : 0);  n = nt*16 + (lane&15)
  const int kparity = (lane >= 16) ? 2 : 0;
  const int ncol = lane & 15;

  float2 breg[16];
  auto fetchB = [&](int kc) {
#pragma unroll
    for (int i = 0; i < 16; ++i) {
      int idx8 = i * 8 + wv;
      int nt = idx8 & 15;
      int ks = idx8 >> 4;
      int k = kc + ks * 4 + kparity;
      const float* bp = Bw + (size_t)k * 256 + nt * 16 + ncol;
      breg[i] = make_float2(bp[0], bp[256]);
    }
  };
  fetchB(0);

  for (int kc = 0; kc < K; kc += 32) {
    __syncthreads();
#pragma unroll
    for (int i = 0; i < 16; ++i) {
      int idx8 = i * 8 + wv;
      lB[idx8 >> 4][idx8 & 15][lane] = breg[i];
    }
    __syncthreads();
    if (kc + 32 < K) fetchB(kc + 32);  // prefetch next chunk (uniform branch)

#pragma unroll
    for (int ks = 0; ks < 8; ++ks) {
      float2 av = *reinterpret_cast<const float2*>(Arow + kc + ks * 4);
      v2f af;
      af.x = av.x;
      af.y = av.y;
      v2f bf[16];
#pragma unroll
      for (int nt = 0; nt < 16; ++nt) {
        float2 bv = lB[ks][nt][lane];
        bf[nt].x = bv.x;
        bf[nt].y = bv.y;
      }
#pragma unroll
      for (int nt = 0; nt < 16; ++nt) {
        acc[nt] = __builtin_amdgcn_wmma_f32_16x16x4_f32(
            false, af, false, bf[nt], (short)0, acc[nt], false, false);
      }
    }
  }

  // Epilogue. C/D layout: vgpr r -> row r (lanes 0-15) / r+8 (lanes 16-31),
  // col = nt*16 + (lane&15)
  const int moff = (lane >> 4) * 8;
#pragma unroll
  for (int nt = 0; nt < 16; ++nt) {
    const int col = nt * 16 + ncol;
    float bv = bias ? bias[col] : 0.0f;
    if (bias2) bv += bias2[col];
#pragma unroll
    for (int r = 0; r < 8; ++r) {
      int row = r0 + moff + r;
      if (row < nrows) {
        float v = acc[nt][r] + bv;
        if (act) v = gelu_tanh(v);
        size_t o = (size_t)row * 256 + col;
        if (Res) v += Res[o];
        Out[o] = v;
      }
    }
  }
}

// -------------------------------------------------------------------------
// LayerNorm over 256 features, one wave per row
// -------------------------------------------------------------------------
__global__ __launch_bounds__(256) void ln256(const float* __restrict__ X,
                                             const float* __restrict__ g,
                                             const float* __restrict__ b,
                                             float* __restrict__ Y, int nrows) {
  int lane = threadIdx.x & 31;
  int wv = threadIdx.x >> 5;
  int row = blockIdx.x * 8 + wv;
  if (row >= nrows) return;
  const float4* xr = reinterpret_cast<const float4*>(X + (size_t)row * 256);
  float4 a = xr[lane];
  float4 c = xr[lane + 32];
  float s = a.x + a.y + a.z + a.w + c.x + c.y + c.z + c.w;
  s = wave_sum(s);
  float mean = s * (1.0f / 256.0f);
  float d0 = a.x - mean, d1 = a.y - mean, d2 = a.z - mean, d3 = a.w - mean;
  float d4 = c.x - mean, d5 = c.y - mean, d6 = c.z - mean, d7 = c.w - mean;
  float vs = d0 * d0 + d1 * d1 + d2 * d2 + d3 * d3 + d4 * d4 + d5 * d5 +
             d6 * d6 + d7 * d7;
  vs = wave_sum(vs);
  float rs = rsqrtf(vs * (1.0f / 256.0f) + 1e-5f);
  const float4* gp = reinterpret_cast<const float4*>(g);
  const float4* bp = reinterpret_cast<const float4*>(b);
  float4 g0 = gp[lane], g1 = gp[lane + 32], b0 = bp[lane], b1 = bp[lane + 32];
  float4 y0, y1;
  y0.x = d0 * rs * g0.x + b0.x;
  y0.y = d1 * rs * g0.y + b0.y;
  y0.z = d2 * rs * g0.z + b0.z;
  y0.w = d3 * rs * g0.w + b0.w;
  y1.x = d4 * rs * g1.x + b1.x;
  y1.y = d5 * rs * g1.y + b1.y;
  y1.z = d6 * rs * g1.z + b1.z;
  y1.w = d7 * rs * g1.w + b1.w;
  float4* yr = reinterpret_cast<float4*>(Y + (size_t)row * 256);
  yr[lane] = y0;
  yr[lane + 32] = y1;
}

// -------------------------------------------------------------------------
// Preprocess stage 1: Hid[n, j<512] = gelu(x[n,:8] @ W1 + b1)
// -------------------------------------------------------------------------
__global__ __launch_bounds__(256) void pre_mlp1(const float* __restrict__ X,
                                                const float* __restrict__ W1,
                                                const float* __restrict__ b1,
                                                float* __restrict__ Hid,
                                                int nrows) {
  int idx = blockIdx.x * blockDim.x + threadIdx.x;
  int n = idx >> 9;
  int j = idx & 511;
  if (n >= nrows) return;
  float a = b1[j];
#pragma unroll
  for (int k = 0; k < 8; ++k) a += X[n * 8 + k] * W1[k * 512 + j];
  Hid[(size_t)n * 512 + j] = gelu_tanh(a);
}

// -------------------------------------------------------------------------
// Slice weights: w[n,h,s] = softmax_s((x_mid[n,h,:] @ Wslice + bslice)/gtemp[h])
// One wave per (node, head); lane = slice index.
// -------------------------------------------------------------------------
__global__ __launch_bounds__(256) void slice_softmax(
    const float* __restrict__ Xmid, const float* __restrict__ Ws,
    const float* __restrict__ bs, const float* __restrict__ gtemp,
    float* __restrict__ Wout, int nrows) {
  int lane = threadIdx.x & 31;
  int wv = threadIdx.x >> 5;
  int flat = blockIdx.x * 8 + wv;
  int n = flat >> 3, h = flat & 7;
  if (n >= nrows) return;
  const float* xp = Xmid + (size_t)n * 256 + h * 32;
  float lg = bs[lane];
#pragma unroll 8
  for (int d = 0; d < 32; ++d) lg += xp[d] * Ws[d * 32 + lane];
  lg = lg / gtemp[h];
  float mx = wave_max(lg);
  float e = expf(lg - mx);
  float sm = wave_sum(e);
  Wout[(size_t)n * 256 + h * 32 + lane] = e / sm;
}

// -------------------------------------------------------------------------
// Graph offsets via binary search (batch is sorted)
// -------------------------------------------------------------------------
__global__ void seg_offsets(const int* __restrict__ batch,
                            int* __restrict__ offs, int n) {
  int t = threadIdx.x;
  if (t > BGRAPH) return;
  int lo = 0, hi = n;
  while (lo < hi) {
    int mid = (lo + hi) >> 1;
    if (batch[mid] < t) lo = mid + 1;
    else hi = mid;
  }
  offs[t] = lo;
}

// -------------------------------------------------------------------------
// Token accumulation: tok[b,h,s,d] = sum_n w[n,h,s]*fx_mid[n,h,d] /
//                                    (sum_n w[n,h,s] + eps)
// One block per (b,h); thread -> (s, 4 d's); atomic-free (batch sorted).
// -------------------------------------------------------------------------
__global__ __launch_bounds__(256) void tok_accum(const float* __restrict__ Wbuf,
                                                 const float* __restrict__ FX,
                                                 const int* __restrict__ offs,
                                                 float* __restrict__ Tok) {
  __shared__ float snorm[32];
  int b = blockIdx.x >> 3, h = blockIdx.x & 7;
  int tid = threadIdx.x;
  int s = tid >> 3, d0 = (tid & 7) * 4;
  int n0 = offs[b], n1 = offs[b + 1];
  float ax = 0.f, ay = 0.f, az = 0.f, aw = 0.f, nrm = 0.f;
  for (int n = n0; n < n1; ++n) {
    float wval = Wbuf[(size_t)n * 256 + h * 32 + s];
    float4 fx =
        *reinterpret_cast<const float4*>(FX + (size_t)n * 256 + h * 32 + d0);
    ax += wval * fx.x;
    ay += wval * fx.y;
    az += wval * fx.z;
    aw += wval * fx.w;
    if ((tid & 7) == 0) nrm += wval;
  }
  if ((tid & 7) == 0) snorm[s] = nrm;
  __syncthreads();
  float inv = 1.0f / (snorm[s] + 1e-5f);
  float* tp = Tok + ((size_t)blockIdx.x * 32 + s) * 32 + d0;
  tp[0] = ax * inv;
  tp[1] = ay * inv;
  tp[2] = az * inv;
  tp[3] = aw * inv;
}

// -------------------------------------------------------------------------
// Per-(b,h) 32x32 slice attention, fully in LDS
// -------------------------------------------------------------------------
__global__ __launch_bounds__(256) void slice_attention(
    const float* __restrict__ Tok, const float* __restrict__ Wq,
    const float* __restrict__ Wk, const float* __restrict__ Wv,
    float* __restrict__ Otok) {
  __shared__ float t[1024], q[1024], kk[1024], vv[1024], at[1024];
  __shared__ float rmax[32], rinv[32];
  int tid = threadIdx.x;
  const float* tp = Tok + (size_t)blockIdx.x * 1024;
#pragma unroll
  for (int i = 0; i < 4; ++i) t[tid * 4 + i] = tp[tid * 4 + i];
  __syncthreads();
  int r = tid >> 3, c0 = (tid & 7) * 4;
#pragma unroll
  for (int j = 0; j < 4; ++j) {
    int c = c0 + j;
    float aq = 0.f, ak = 0.f, av = 0.f;
    for (int d = 0; d < 32; ++d) {
      float tv = t[r * 32 + d];
      aq += tv * Wq[d * 32 + c];
      ak += tv * Wk[d * 32 + c];
      av += tv * Wv[d * 32 + c];
    }
    q[r * 32 + c] = aq;
    kk[r * 32 + c] = ak;
    vv[r * 32 + c] = av;
  }
  __syncthreads();
#pragma unroll
  for (int j = 0; j < 4; ++j) {
    int c = c0 + j;
    float dt = 0.f;
    for (int d = 0; d < 32; ++d) dt += q[r * 32 + d] * kk[c * 32 + d];
    at[r * 32 + c] = dt * 0.17677669529663687f;  // 32^-0.5
  }
  __syncthreads();
  if (tid < 32) {
    float m = -1e30f;
    for (int c = 0; c < 32; ++c) m = fmaxf(m, at[tid * 32 + c]);
    float sum = 0.f;
    for (int c = 0; c < 32; ++c) sum += expf(at[tid * 32 + c] - m);
    rmax[tid] = m;
    rinv[tid] = 1.0f / sum;
  }
  __syncthreads();
#pragma unroll
  for (int j = 0; j < 4; ++j) {
    int c = c0 + j;
    at[r * 32 + c] = expf(at[r * 32 + c] - rmax[r]) * rinv[r];
  }
  __syncthreads();
  float* op = Otok + (size_t)blockIdx.x * 1024;
#pragma unroll
  for (int j = 0; j < 4; ++j) {
    int c = c0 + j;
    float o = 0.f;
    for (int s = 0; s < 32; ++s) o += at[r * 32 + s] * vv[s * 32 + c];
    op[r * 32 + c] = o;
  }
}

// -------------------------------------------------------------------------
// De-slice: out_x[n,h,d] = sum_s w[n,h,s] * otok[batch[n],h,s,d]
// One block per node; thread -> (h,d).
// -------------------------------------------------------------------------
__global__ __launch_bounds__(256) void deslice(const float* __restrict__ Wbuf,
                                               const float* __restrict__ Otok,
                                               const int* __restrict__ batch,
                                               float* __restrict__ OutX,
                                               int nrows) {
  int n = blockIdx.x;
  int h = threadIdx.x >> 5, d = threadIdx.x & 31;
  int b = batch[n];
  const float* wp = Wbuf + (size_t)n * 256 + h * 32;
  const float* op = Otok + (size_t)(b * 8 + h) * 1024 + d;
  float a = 0.f;
#pragma unroll 8
  for (int s = 0; s < 32; ++s) a += wp[s] * op[s * 32];
  OutX[(size_t)n * 256 + h * 32 + d] = a;
}

// -------------------------------------------------------------------------
// Final decode: Out[n,:4] = X[n,:] @ Wd + bd
// -------------------------------------------------------------------------
__global__ __launch_bounds__(256) void decode_out(const float* __restrict__ X,
                                                  const float* __restrict__ Wd,
                                                  const float* __restrict__ bd,
                                                  float* __restrict__ Out,
                                                  int nrows) {
  int n = blockIdx.x * blockDim.x + threadIdx.x;
  if (n >= nrows) return;
  float a0 = bd[0], a1 = bd[1], a2 = bd[2], a3 = bd[3];
  const float* xp = X + (size_t)n * 256;
  for (int k = 0; k < 256; ++k) {
    float xv = xp[k];
    float4 wr = *reinterpret_cast<const float4*>(Wd + k * 4);
    a0 += xv * wr.x;
    a1 += xv * wr.y;
    a2 += xv * wr.z;
    a3 += xv * wr.w;
  }
  float4 o = make_float4(a0, a1, a2, a3);
  *reinterpret_cast<float4*>(Out + (size_t)n * 4) = o;
}

// -------------------------------------------------------------------------
// Host orchestration
// -------------------------------------------------------------------------
struct Blk {
  const float *Wfx, *Wk, *Wo, *Wq, *Wslice, *Wv, *Wx;
  const float *bfx, *bo, *bslice, *bx, *gtemp;
  const float *ln1b, *ln1g, *ln2b, *ln2g;
  const float *mW1, *mW2, *mb1, *mb2;
};

extern "C" void kernel_launch(void* const* d_in, const int* in_sizes, int n_in,
                              void* d_out, int out_size, void* d_ws,
                              size_t ws_size, hipStream_t stream) {
  auto P = [&](int i) { return (const float*)d_in[i]; };

  // Detect flattening order: insertion order has x (N*8) first, batch (N)
  // second; alphabetical tree_leaves order has batch first.
  const bool insertion = (in_sizes[0] == 8 * in_sizes[1]);

  const int* batch;
  const float *x, *preW1, *preb1, *preW2, *preb2, *placeholder;
  const float *decW, *decb, *decbW, *decg;
  Blk blk[NLAYERS];

  if (!insertion) {
    // alphabetical (jax tree_leaves): batch, blocks[...], dec_W, dec_b,
    // dec_bW, dec_g, placeholder, pre_W1, pre_W2, pre_b1, pre_b2, x
    batch = (const int*)d_in[0];
    for (int L = 0; L < NLAYERS; ++L) {
      int o = 1 + L * 20;
      blk[L].Wfx = P(o + 0);
      blk[L].Wk = P(o + 1);
      blk[L].Wo = P(o + 2);
      blk[L].Wq = P(o + 3);
      blk[L].Wslice = P(o + 4);
      blk[L].Wv = P(o + 5);
      blk[L].Wx = P(o + 6);
      blk[L].bfx = P(o + 7);
      blk[L].bo = P(o + 8);
      blk[L].bslice = P(o + 9);
      blk[L].bx = P(o + 10);
      blk[L].gtemp = P(o + 11);
      blk[L].ln1b = P(o + 12);
      blk[L].ln1g = P(o + 13);
      blk[L].ln2b = P(o + 14);
      blk[L].ln2g = P(o + 15);
      blk[L].mW1 = P(o + 16);
      blk[L].mW2 = P(o + 17);
      blk[L].mb1 = P(o + 18);
      blk[L].mb2 = P(o + 19);
    }
    decW = P(101);
    decb = P(102);
    decbW = P(103);
    decg = P(104);
    placeholder = P(105);
    preW1 = P(106);
    preW2 = P(107);
    preb1 = P(108);
    preb2 = P(109);
    x = P(110);
  } else {
    // insertion order of the dict literals in setup_inputs()
    x = P(0);
    batch = (const int*)d_in[1];
    preW1 = P(2);
    preb1 = P(3);
    preW2 = P(4);
    preb2 = P(5);
    placeholder = P(6);
    for (int L = 0; L < NLAYERS; ++L) {
      int o = 7 + L * 20;
      blk[L].ln1g = P(o + 0);
      blk[L].ln1b = P(o + 1);
      blk[L].Wx = P(o + 2);
      blk[L].bx = P(o + 3);
      blk[L].Wfx = P(o + 4);
      blk[L].bfx = P(o + 5);
      blk[L].Wslice = P(o + 6);
      blk[L].bslice = P(o + 7);
      blk[L].gtemp = P(o + 8);
      blk[L].Wq = P(o + 9);
      blk[L].Wk = P(o + 10);
      blk[L].Wv = P(o + 11);
      blk[L].Wo = P(o + 12);
      blk[L].bo = P(o + 13);
      blk[L].ln2g = P(o + 14);
      blk[L].ln2b = P(o + 15);
      blk[L].mW1 = P(o + 16);
      blk[L].mb1 = P(o + 17);
      blk[L].mW2 = P(o + 18);
      blk[L].mb2 = P(o + 19);
    }
    decg = P(107);
    decb = P(108);
    decW = P(109);
    decbW = P(110);
  }

  const int N = insertion ? in_sizes[1] : in_sizes[0];

  // Workspace layout (256B-aligned slabs)
  char* wsp = (char*)d_ws;
  const size_t NB = (size_t)N * 256 * sizeof(float);
  float* fxA = (float*)(wsp);           // residual stream
  float* S1 = (float*)(wsp + NB);       // ln out / slice weights w
  float* S2 = (float*)(wsp + 2 * NB);   // x_mid / out_x / mlp hidden
  float* S3 = (float*)(wsp + 3 * NB);   // fx_mid
  float* tok = (float*)(wsp + 4 * NB);  // [B,H,S,D]
  float* otok = tok + BGRAPH * HEADS * SLICES * DHEAD;
  int* offs = (int*)(otok + BGRAPH * HEADS * SLICES * DHEAD);
  float* hidden = S2;  // N*512 floats: spans S2..S3 (pre-phase only)

  const int gemmGrid = (N + 127) / 128;
  const int lnGrid = (N + 7) / 8;

  // Preprocess MLP + placeholder
  pre_mlp1<<<(N * 512 + 255) / 256, 256, 0, stream>>>(x, preW1, preb1, hidden,
                                                      N);
  wmma_gemm_n256<<<gemmGrid, 256, 0, stream>>>(hidden, preW2, preb2,
                                               placeholder, nullptr, fxA, N,
                                               PREHID, 0);
  seg_offsets<<<1, 32, 0, stream>>>(batch, offs, N);

  for (int L = 0; L < NLAYERS; ++L) {
    const Blk& p = blk[L];
    // attention branch
    ln256<<<lnGrid, 256, 0, stream>>>(fxA, p.ln1g, p.ln1b, S1, N);
    wmma_gemm_n256<<<gemmGrid, 256, 0, stream>>>(S1, p.Wx, p.bx, nullptr,
                                                 nullptr, S2, N, 256, 0);
    wmma_gemm_n256<<<gemmGrid, 256, 0, stream>>>(S1, p.Wfx, p.bfx, nullptr,
                                                 nullptr, S3, N, 256, 0);
    slice_softmax<<<N, 256, 0, stream>>>(S2, p.Wslice, p.bslice, p.gtemp, S1,
                                         N);
    tok_accum<<<BGRAPH * HEADS, 256, 0, stream>>>(S1, S3, offs, tok);
    slice_attention<<<BGRAPH * HEADS, 256, 0, stream>>>(tok, p.Wq, p.Wk, p.Wv,
                                                        otok);
    deslice<<<N, 256, 0, stream>>>(S1, otok, batch, S2, N);
    wmma_gemm_n256<<<gemmGrid, 256, 0, stream>>>(S2, p.Wo, p.bo, nullptr, fxA,
                                                 fxA, N, 256, 0);
    // MLP branch
    ln256<<<lnGrid, 256, 0, stream>>>(fxA, p.ln2g, p.ln2b, S1, N);
    wmma_gemm_n256<<<gemmGrid, 256, 0, stream>>>(S1, p.mW1, p.mb1, nullptr,
                                                 nullptr, S2, N, 256, 1);
    wmma_gemm_n256<<<gemmGrid, 256, 0, stream>>>(S2, p.mW2, p.mb2, nullptr, fxA,
                                                 fxA, N, 256, 0);
  }

  // Decoder
  ln256<<<lnGrid, 256, 0, stream>>>(fxA, decg, decb, S1, N);
  decode_out<<<(N + 255) / 256, 256, 0, stream>>>(S1, decW, decbW,
                                                  (float*)d_out, N);
}